// APPNPModel_24318104830502
// MI455X (gfx1250) — compile-verified
//
#include <hip/hip_runtime.h>
#include <math.h>

#define N_NODES  100000
#define HIDDEN   64
#define N_LABELS 40
#define ALPHA    0.1f
#define ITERS    10

typedef __attribute__((ext_vector_type(2))) float v2f;
typedef __attribute__((ext_vector_type(8))) float v8f;

// ---------------- utility ----------------
__global__ void fill_zero_f32(float* __restrict__ p, int n) {
    int i = blockIdx.x * blockDim.x + threadIdx.x;
    if (i < n) p[i] = 0.0f;
}

// ---------------- stage 1: feature SpMM ----------------
// one wave32 per nnz; each lane handles 2 consecutive hidden channels (64 total)
__global__ void spmm_feat(const int* __restrict__ rows, const int* __restrict__ cols,
                          const float* __restrict__ vals, const float* __restrict__ W1,
                          float* __restrict__ h1, int nnz) {
    long tid = (long)blockIdx.x * blockDim.x + threadIdx.x;
    int e    = (int)(tid >> 5);
    int lane = (int)(tid & 31);
    if (e >= nnz) return;
    int   r = rows[e];
    int   c = cols[e];
    float v = vals[e];
    const float2* w  = (const float2*)(W1 + (long)c * HIDDEN + lane * 2);
    float2 wv = *w;
    float* out = h1 + (long)r * HIDDEN + lane * 2;
    atomicAdd(out + 0, v * wv.x);
    atomicAdd(out + 1, v * wv.y);
}

// h1 = relu(h1 + b1), in place
__global__ void relu_bias(float* __restrict__ h1, const float* __restrict__ b1, int n) {
    int i = blockIdx.x * blockDim.x + threadIdx.x;
    if (i < n) {
        float x = h1[i] + b1[i & (HIDDEN - 1)];
        h1[i] = x > 0.0f ? x : 0.0f;
    }
}

// ---------------- stage 2: dense GEMM via fp32 WMMA ----------------
// h2[100000,40] = h1[100000,64] @ W2[64,40] + b2
// One wave32 per TWO 16x16 output tiles (B fragments register-resident, reused).
// K=64 -> 16 chained V_WMMA_F32_16X16X4_F32 per tile.
#define MTILES (N_NODES / 16)   // 6250 (exact, even)

__global__ void gemm_h2_wmma(const float* __restrict__ h1, const float* __restrict__ W2,
                             const float* __restrict__ b2, float* __restrict__ h2) {
    const int wave = threadIdx.x >> 5;
    const int lane = threadIdx.x & 31;
    const int wid  = blockIdx.x * 4 + wave;           // wave id within this ntile
    const int n0   = blockIdx.y * 16;                 // 3 col tiles (last partial)
    const int mtile0 = wid * 2;
    if (mtile0 >= MTILES) return;                     // wave-uniform guard (EXEC all-1s)

    const int half = lane >> 4;                       // 0: K+{0,1}, 1: K+{2,3}
    const int l15  = lane & 15;
    const int bcol = n0 + l15;                        // B fragment: lane -> col N

    // ---- preload all 16 B fragments, branch-free (clamp + mask) ----
    const int   bc    = bcol < (N_LABELS - 1) ? bcol : (N_LABELS - 1);
    const float bmask = (bcol < N_LABELS) ? 1.0f : 0.0f;
    v2f bfrag[16];
#pragma unroll
    for (int kk = 0; kk < 16; ++kk) {
        const int ka = kk * 4 + half * 2;
        bfrag[kk].x = W2[(ka)     * N_LABELS + bc] * bmask;
        bfrag[kk].y = W2[(ka + 1) * N_LABELS + bc] * bmask;
    }
    const float bias = (bcol < N_LABELS) ? b2[bc] : 0.0f;

    // ---- two M tiles per wave, reusing bfrag ----
#pragma unroll
    for (int t = 0; t < 2; ++t) {
        const int m0   = (mtile0 + t) * 16;
        const int arow = m0 + l15;                    // A fragment: lane -> row M
        const float* ap = h1 + (long)arow * HIDDEN + half * 2;

        v8f acc = {};
#pragma unroll
        for (int kk = 0; kk < 16; ++kk) {
            v2f a = *(const v2f*)(ap + kk * 4);       // global_load_b64, contiguous pair
            acc = __builtin_amdgcn_wmma_f32_16x16x4_f32(
                /*neg_a=*/false, a, /*neg_b=*/false, bfrag[kk],
                /*c_mod=*/(short)0, acc, /*reuse_a=*/false, /*reuse_b=*/false);
        }
        // D layout: VGPR r -> M = m0 + r + 8*half, N = n0 + l15
#pragma unroll
        for (int r = 0; r < 8; ++r) {
            const int row = m0 + r + half * 8;
            if (bcol < N_LABELS)
                h2[(long)row * N_LABELS + bcol] = acc[r] + bias;
        }
    }
}

// ---------------- stage 3: APPNP propagation ----------------
// one wave32 per edge; lanes cover the 40 label channels (lanes 0-7 take 2)
__global__ void spmm_edge(const int* __restrict__ er, const int* __restrict__ ec,
                          const float* __restrict__ ew, const float* __restrict__ src,
                          float* __restrict__ acc, int nedges) {
    long tid = (long)blockIdx.x * blockDim.x + threadIdx.x;
    int e    = (int)(tid >> 5);
    int lane = (int)(tid & 31);
    if (e >= nedges) return;
    int   r = er[e];
    int   c = ec[e];
    float w = ew[e];
    const float* s = src + (long)c * N_LABELS;
    float*       d = acc + (long)r * N_LABELS;
    atomicAdd(d + lane, w * s[lane]);
    if (lane < N_LABELS - 32)
        atomicAdd(d + 32 + lane, w * s[32 + lane]);
}

// p = tanh((1-alpha)*acc + alpha*h2)
__global__ void appnp_update(const float* __restrict__ acc, const float* __restrict__ h2,
                             float* __restrict__ p, int n) {
    int i = blockIdx.x * blockDim.x + threadIdx.x;
    if (i < n) p[i] = tanhf((1.0f - ALPHA) * acc[i] + ALPHA * h2[i]);
}

// ---------------- stage 4: row-wise log_softmax ----------------
// one wave32 per row of 40
__global__ void log_softmax_rows(const float* __restrict__ p, float* __restrict__ out) {
    long tid = (long)blockIdx.x * blockDim.x + threadIdx.x;
    int row  = (int)(tid >> 5);
    int lane = (int)(tid & 31);
    if (row >= N_NODES) return;
    const float* x = p + (long)row * N_LABELS;
    float v0 = x[lane];
    float v1 = (lane < N_LABELS - 32) ? x[32 + lane] : -INFINITY;
    float m = fmaxf(v0, v1);
#pragma unroll
    for (int off = 16; off > 0; off >>= 1)
        m = fmaxf(m, __shfl_xor(m, off, 32));
    float s = __expf(v0 - m) + ((lane < N_LABELS - 32) ? __expf(v1 - m) : 0.0f);
#pragma unroll
    for (int off = 16; off > 0; off >>= 1)
        s += __shfl_xor(s, off, 32);
    float lse = m + __logf(s);
    float* o = out + (long)row * N_LABELS;
    o[lane] = v0 - lse;
    if (lane < N_LABELS - 32)
        o[32 + lane] = v1 - lse;
}

// ---------------- host-side orchestration ----------------
extern "C" void kernel_launch(void* const* d_in, const int* in_sizes, int n_in,
                              void* d_out, int out_size, void* d_ws, size_t ws_size,
                              hipStream_t stream) {
    const int*   feature_rows = (const int*)  d_in[0];
    const int*   feature_cols = (const int*)  d_in[1];
    const float* feature_vals = (const float*)d_in[2];
    const int*   edge_rows    = (const int*)  d_in[3];
    const int*   edge_cols    = (const int*)  d_in[4];
    const float* edge_weights = (const float*)d_in[5];
    const float* W1           = (const float*)d_in[6];
    const float* b1           = (const float*)d_in[7];
    const float* W2           = (const float*)d_in[8];
    const float* b2           = (const float*)d_in[9];
    float*       out          = (float*)d_out;

    const int nnz    = in_sizes[0];
    const int nedges = in_sizes[3];

    // workspace layout (all offsets 256B-aligned)
    char*  ws  = (char*)d_ws;
    float* h1  = (float*)(ws);                                   // 100000*64*4  = 25,600,000 B
    float* h2  = (float*)(ws + 25600000);                        // 100000*40*4  = 16,000,000 B
    float* p   = (float*)(ws + 25600000 + 16000000);             // 16,000,000 B
    float* acc = (float*)(ws + 25600000 + 2 * 16000000);         // 16,000,000 B

    const int n_h1 = N_NODES * HIDDEN;    // 6,400,000
    const int n_h2 = N_NODES * N_LABELS;  // 4,000,000

    // stage 1: h1 = relu(spmm(X, W1) + b1)
    fill_zero_f32<<<(n_h1 + 255) / 256, 256, 0, stream>>>(h1, n_h1);
    {
        long threads = (long)nnz * 32;
        spmm_feat<<<(int)((threads + 255) / 256), 256, 0, stream>>>(
            feature_rows, feature_cols, feature_vals, W1, h1, nnz);
    }
    relu_bias<<<(n_h1 + 255) / 256, 256, 0, stream>>>(h1, b1, n_h1);

    // stage 2: h2 = h1 @ W2 + b2   (fp32 WMMA, 2 M-tiles per wave)
    {
        const int waves_per_ntile = (MTILES / 2) + 3;   // 3125 waves needed, round up to blocks of 4
        dim3 grid(waves_per_ntile / 4, (N_LABELS + 15) / 16);
        gemm_h2_wmma<<<grid, 128, 0, stream>>>(h1, W2, b2, h2);
    }

    // stage 3: APPNP loop; iteration 0 propagates h2 directly, then p in place
    for (int it = 0; it < ITERS; ++it) {
        fill_zero_f32<<<(n_h2 + 255) / 256, 256, 0, stream>>>(acc, n_h2);
        const float* src = (it == 0) ? h2 : p;
        long threads = (long)nedges * 32;
        spmm_edge<<<(int)((threads + 255) / 256), 256, 0, stream>>>(
            edge_rows, edge_cols, edge_weights, src, acc, nedges);
        appnp_update<<<(n_h2 + 255) / 256, 256, 0, stream>>>(acc, h2, p, n_h2);
    }

    // stage 4: out = log_softmax(p, axis=1)
    {
        long threads = (long)N_NODES * 32;
        log_softmax_rows<<<(int)((threads + 255) / 256), 256, 0, stream>>>(p, out);
    }
}